// CRPSLoss_51625506898386
// MI455X (gfx1250) — compile-verified
//
#include <hip/hip_runtime.h>

// ---------------------------------------------------------------------------
// CRPS (empirical, mean reduction) for forecasts (N=20, M points) vs target.
//   crps_p = (1/N) * sum_i |x_i - y|  -  (1/N^2) * sum_{i<j} |x_i - x_j|
//   out    = mean_p crps_p
// Bandwidth-bound: 264 MB @ 23.3 TB/s ~ 11.3 us. CDNA5 path: TDM
// tensor_load_to_lds DMAs the (N x TILE) strided tile (row stride = M elems)
// into LDS in a single descriptor; waves consume via ds_load_b64.
// ---------------------------------------------------------------------------

#define NSAMP 20
#define TILE  512
#define BLOCK 256
#define PPT   (TILE / BLOCK)   // 2 points per thread

typedef unsigned int v4u __attribute__((ext_vector_type(4)));
typedef int          v8i __attribute__((ext_vector_type(8)));
typedef int          v4i __attribute__((ext_vector_type(4)));

__device__ __forceinline__ float crps_point(const float* x, float y) {
  float t1 = 0.0f, pw = 0.0f;
#pragma unroll
  for (int i = 0; i < NSAMP; ++i) t1 += __builtin_fabsf(x[i] - y);
#pragma unroll
  for (int i = 0; i < NSAMP; ++i)
#pragma unroll
    for (int j = i + 1; j < NSAMP; ++j)
      pw += __builtin_fabsf(x[i] - x[j]);
  return t1 * (1.0f / NSAMP) - pw * (1.0f / (NSAMP * NSAMP));
}

__device__ __forceinline__ float wave_reduce_add(float v) {
#pragma unroll
  for (int o = 16; o > 0; o >>= 1) v += __shfl_xor(v, o, 32);
  return v;
}

__global__ __launch_bounds__(BLOCK)
void crps_tile_kernel(const float* __restrict__ forecasts,
                      const float* __restrict__ target,
                      float* __restrict__ partial,
                      unsigned long long sampleStride /* M elements */) {
  __shared__ float tileBuf[NSAMP * TILE];   // 40 KB staged tile
  __shared__ float redBuf[BLOCK / 32];

  const unsigned tid = threadIdx.x;
  const unsigned long long base = (unsigned long long)blockIdx.x * (unsigned long long)TILE;

  // --- One TDM op per workgroup: 2D tile (tile_dim0=TILE cols, tile_dim1=NSAMP
  // rows, row stride = M elements) -> contiguous LDS. Issued by wave 0 only
  // (TDM ignores EXEC; one instruction per wave = one DMA descriptor).
  if (tid < 32) {
    const unsigned long long gaddr =
        (unsigned long long)forecasts + base * 4ull;
    const unsigned ldsOff = (unsigned)(unsigned long long)&tileBuf[0]; // LDS byte addr (low 32 bits of flat)

    // D# group 0 (128b): count=1 | lds_addr | global_addr[56:0] | type=2
    v4u g0;
    g0.x = 1u;
    g0.y = ldsOff;
    g0.z = (unsigned)gaddr;
    g0.w = ((unsigned)(gaddr >> 32) & 0x01FFFFFFu) | 0x80000000u;

    // D# group 1 (256b)
    v8i g1;
    g1[0] = (int)(2u << 16);                         // data_size = 4 bytes; no mask/pad/iterate
    g1[1] = (int)((unsigned)TILE << 16);             // tensor_dim0[15:0] (bits 63:48)
    g1[2] = (int)((unsigned)NSAMP << 16);            // tensor_dim0[31:16]=0 | tensor_dim1[15:0]
    g1[3] = (int)((unsigned)TILE << 16);             // tensor_dim1[31:16]=0 | tile_dim0
    g1[4] = (int)(unsigned)NSAMP;                    // tile_dim1 | tile_dim2=0 (2D)
    g1[5] = (int)(unsigned)sampleStride;             // tensor_dim0_stride[31:0] = M
    g1[6] = (int)((unsigned)(sampleStride >> 32) & 0xFFFFu); // stride0 hi | stride1 lo = 0
    g1[7] = 0;

    v4i gz4 = {0, 0, 0, 0};                          // groups 2/3: unused (2D tile)
    v8i gz8 = {0, 0, 0, 0, 0, 0, 0, 0};              // extra group (clang-23 6-arg form): zeros
    __builtin_amdgcn_tensor_load_to_lds(g0, g1, gz4, gz4, gz8, 0);
    __builtin_amdgcn_s_wait_tensorcnt(0);
  }
  __syncthreads();

  // --- Per-thread compute: 2 points, 20 samples each from LDS (ds_load_b64).
  const unsigned p = tid * PPT;
  float x0[NSAMP], x1[NSAMP];
#pragma unroll
  for (int i = 0; i < NSAMP; ++i) {
    const float2 v = *(const float2*)&tileBuf[i * TILE + p];
    x0[i] = v.x;
    x1[i] = v.y;
  }
  const float2 y2 = *(const float2*)&target[base + p];

  float acc = crps_point(x0, y2.x) + crps_point(x1, y2.y);

  // --- Deterministic block reduction (wave32 shuffles + LDS).
  acc = wave_reduce_add(acc);
  if ((tid & 31u) == 0u) redBuf[tid >> 5] = acc;
  __syncthreads();
  if (tid < BLOCK / 32) {
    float v = redBuf[tid];
#pragma unroll
    for (int o = BLOCK / 64; o > 0; o >>= 1) v += __shfl_xor(v, o, 32);
    if (tid == 0) partial[blockIdx.x] = v;
  }
}

// Fallback for a ragged tail (M % TILE != 0) — plain coalesced loads.
__global__ __launch_bounds__(BLOCK)
void crps_tail_kernel(const float* __restrict__ forecasts,
                      const float* __restrict__ target,
                      float* __restrict__ partial,
                      unsigned long long start,
                      unsigned long long M,
                      int slot) {
  __shared__ float redBuf[BLOCK / 32];
  float acc = 0.0f;
  for (unsigned long long idx = start + threadIdx.x; idx < M; idx += BLOCK) {
    float x[NSAMP];
#pragma unroll
    for (int i = 0; i < NSAMP; ++i) x[i] = forecasts[(unsigned long long)i * M + idx];
    acc += crps_point(x, target[idx]);
  }
  acc = wave_reduce_add(acc);
  if ((threadIdx.x & 31u) == 0u) redBuf[threadIdx.x >> 5] = acc;
  __syncthreads();
  if (threadIdx.x < BLOCK / 32) {
    float v = redBuf[threadIdx.x];
#pragma unroll
    for (int o = BLOCK / 64; o > 0; o >>= 1) v += __shfl_xor(v, o, 32);
    if (threadIdx.x == 0) partial[slot] = v;
  }
}

__global__ __launch_bounds__(256)
void crps_final_reduce(const float* __restrict__ partial, int nPartials,
                       float* __restrict__ out, float invM) {
  __shared__ float redBuf[8];
  float acc = 0.0f;
  for (int i = threadIdx.x; i < nPartials; i += 256) acc += partial[i];
  acc = wave_reduce_add(acc);
  if ((threadIdx.x & 31u) == 0u) redBuf[threadIdx.x >> 5] = acc;
  __syncthreads();
  if (threadIdx.x < 8) {
    float v = redBuf[threadIdx.x];
#pragma unroll
    for (int o = 4; o > 0; o >>= 1) v += __shfl_xor(v, o, 32);
    if (threadIdx.x == 0) out[0] = v * invM;
  }
}

extern "C" void kernel_launch(void* const* d_in, const int* in_sizes, int n_in,
                              void* d_out, int out_size, void* d_ws, size_t ws_size,
                              hipStream_t stream) {
  const float* forecasts = (const float*)d_in[0];  // (N=20, M) fp32
  const float* target    = (const float*)d_in[1];  // (M) fp32
  const unsigned long long M = (unsigned long long)in_sizes[1];  // B*C*D*H*W
  const unsigned long long nFull = M / TILE;       // 6144 for the reference shapes
  const unsigned long long rem   = M - nFull * TILE;

  float* partial = (float*)d_ws;
  int nPartials = (int)nFull;

  if (nFull)
    crps_tile_kernel<<<(unsigned)nFull, BLOCK, 0, stream>>>(forecasts, target,
                                                            partial, M);
  if (rem) {
    crps_tail_kernel<<<1, BLOCK, 0, stream>>>(forecasts, target, partial,
                                              nFull * TILE, M, nPartials);
    ++nPartials;
  }
  crps_final_reduce<<<1, 256, 0, stream>>>(partial, nPartials, (float*)d_out,
                                           (float)(1.0 / (double)M));
}